// phi_etakappa_mu_Layer_69578470195648
// MI455X (gfx1250) — compile-verified
//
#include <hip/hip_runtime.h>
#include <stdint.h>

typedef __attribute__((ext_vector_type(4))) float v4f;

#define TPB 256
#define VEC 4
#define TILE_ELEMS (TPB * VEC)            // 1024 floats per tile per stream (4 KiB)
#define R_TILES 16
#define R_ELEMS (TILE_ELEMS * R_TILES)    // 16384 elements handled per reduce block
#define NPB (64 * 256 * 256)              // 4194304 elements per batch
#define R_BLOCKS (NPB / R_ELEMS)          // 256 reduce blocks per batch (== TPB)
#define NBATCH 4

// ---------------------------------------------------------------------------
// helpers
// ---------------------------------------------------------------------------

// sin(pi*x) for |x| <= 1, via fold sin(pi*x) = sign(x)*sin(pi*min(|x|,1-|x|)).
// Polynomial arg s = pi*t with |t| <= 0.5 -> |s| <= pi/2; degree-11 odd poly s + s^5*p,
// ~10 VALU ops, no libm.
__device__ __forceinline__ float sinpi_poly(float x) {
  const float PI = 3.14159265358979323846f;
  float ax = __builtin_fabsf(x);
  float t = (ax > 0.5f) ? (1.0f - ax) : ax;   // sin(pi*ax) == sin(pi*t)
  float s = PI * t;
  float s2 = s * s;
  // s - s^5/6 + s^7/120 - s^9/5040 + s^11/362880
  float p = __builtin_fmaf(s2, 2.7557319e-06f, -1.9841270e-04f);
  p = __builtin_fmaf(s2, p, 8.3333333e-03f);
  p = __builtin_fmaf(s2, p, -1.6666667e-01f);
  p = s * __builtin_fmaf(s2, p * s2, 1.0f);
  return __builtin_copysignf(p, x);
}

__device__ __forceinline__ float heaviside1(float z) {
  // eps = EPSILON * EPSILON_P^it = 1.0 always (EPSILON==EPSILON_P==1)
  const float INV_PI = 0.31830988618379067154f;
  float H = 0.5f * (1.0f + z + sinpi_poly(z) * INV_PI);
  H = (z > 1.0f) ? 1.0f : H;
  H = (z < -1.0f) ? 0.0f : H;
  return H;
}

// deterministic in-block tree reduction (fixed order -> bitwise stable)
__device__ __forceinline__ float block_reduce(float v, float* red) {
  const int tid = threadIdx.x;
  red[tid] = v;
  __syncthreads();
  for (int s = TPB / 2; s > 0; s >>= 1) {
    if (tid < s) red[tid] += red[tid + s];
    __syncthreads();
  }
  float r = red[0];
  __syncthreads();
  return r;
}

// CDNA5 async global -> LDS copy (ASYNCcnt-tracked).  Each lane copies 16B
// into its own LDS slot, so only per-wave counters are needed for sync.
__device__ __forceinline__ void async_copy_b128(uint32_t lds_byte_addr,
                                                const void* gaddr) {
  asm volatile("global_load_async_to_lds_b128 %0, %1, off"
               :
               : "v"(lds_byte_addr), "v"((unsigned long long)(uintptr_t)gaddr)
               : "memory");
}

// ---------------------------------------------------------------------------
// pass 1: per-block partial sums of {H, p*H, p}, async-pipelined through LDS
// ---------------------------------------------------------------------------
__global__ void __launch_bounds__(TPB)
reduce_pass(const float* __restrict__ phi0, const float* __restrict__ pb,
            float* __restrict__ partials) {
  __shared__ __align__(16) float sPhi[2][TILE_ELEMS];
  __shared__ __align__(16) float sPb[2][TILE_ELEMS];
  __shared__ float red[TPB];

  const int tid = threadIdx.x;
  const int b = blockIdx.y;
  const size_t base0 =
      (size_t)b * NPB + (size_t)blockIdx.x * R_ELEMS + (size_t)tid * VEC;

  uint32_t lphi[2], lpb[2];
  lphi[0] = (uint32_t)(uintptr_t)&sPhi[0][tid * VEC];
  lphi[1] = (uint32_t)(uintptr_t)&sPhi[1][tid * VEC];
  lpb[0]  = (uint32_t)(uintptr_t)&sPb[0][tid * VEC];
  lpb[1]  = (uint32_t)(uintptr_t)&sPb[1][tid * VEC];

  auto issue_tile = [&](int t, int buf) {
    // make sure this wave's earlier LDS reads of this buffer are done
    asm volatile("s_wait_dscnt 0" ::: "memory");
    const float* gp = phi0 + base0 + (size_t)t * TILE_ELEMS;
    const float* gq = pb   + base0 + (size_t)t * TILE_ELEMS;
    async_copy_b128(lphi[buf], gp);   // ASYNCcnt += 1 (per wave)
    async_copy_b128(lpb[buf],  gq);   // ASYNCcnt += 1
  };

  float sH = 0.f, sPH = 0.f, sP = 0.f;

  issue_tile(0, 0);
  for (int t = 0; t < R_TILES; ++t) {
    const int buf = t & 1;
    if (t + 1 < R_TILES) {
      issue_tile(t + 1, (t + 1) & 1);
      // 4 outstanding -> wait until only the newest tile's 2 remain
      asm volatile("s_wait_asynccnt 2" ::: "memory");
    } else {
      asm volatile("s_wait_asynccnt 0" ::: "memory");
    }
    v4f z4 = *(const v4f*)&sPhi[buf][tid * VEC];
    v4f p4 = *(const v4f*)&sPb[buf][tid * VEC];
#pragma unroll
    for (int j = 0; j < 4; ++j) {
      const float H = heaviside1(z4[j]);
      sH  += H;
      sPH += p4[j] * H;
      sP  += p4[j];
    }
  }

  sH  = block_reduce(sH, red);
  sPH = block_reduce(sPH, red);
  sP  = block_reduce(sP, red);

  if (tid == 0) {
    const size_t o = ((size_t)b * R_BLOCKS + blockIdx.x) * 3;
    partials[o + 0] = sH;
    partials[o + 1] = sPH;
    partials[o + 2] = sP;
  }
}

// ---------------------------------------------------------------------------
// pass 2: per-batch deterministic reduce of partials -> mu0, mu1
// ---------------------------------------------------------------------------
__global__ void __launch_bounds__(TPB)
mu_pass(const float* __restrict__ partials, float* __restrict__ mu) {
  __shared__ float red[TPB];
  const int b = blockIdx.x;
  const int tid = threadIdx.x;
  const float* p = partials + (size_t)b * R_BLOCKS * 3;  // R_BLOCKS == TPB

  float sH  = p[tid * 3 + 0];
  float sPH = p[tid * 3 + 1];
  float sP  = p[tid * 3 + 2];

  sH  = block_reduce(sH, red);
  sPH = block_reduce(sPH, red);
  sP  = block_reduce(sP, red);

  if (tid == 0) {
    const float mu1 = sPH / sH;                          // sum(p*H)/sum(H)
    const float mu0 = (sP - sPH) / ((float)NPB - sH);    // sum(p*(1-H))/sum(1-H)
    mu[b * 2 + 0] = mu0;
    mu[b * 2 + 1] = mu1;
  }
}

// ---------------------------------------------------------------------------
// pass 3: phi = phi0 + eta*(kappa + (p-mu0)^2 - (p-mu1)^2)
// phi0/pb re-read (L2-resident from pass 1); kappa + output non-temporal.
// ---------------------------------------------------------------------------
__global__ void __launch_bounds__(TPB)
phi_pass(const float* __restrict__ kappa, const float* __restrict__ phi0,
         const float* __restrict__ pb, const float* __restrict__ eta,
         const float* __restrict__ mu, float* __restrict__ out) {
  const int b = blockIdx.y;
  const size_t i4 =
      (size_t)b * (NPB / VEC) + (size_t)blockIdx.x * TPB + threadIdx.x;

  const v4f k = __builtin_nontemporal_load((const v4f*)kappa + i4);
  const v4f z = ((const v4f*)phi0)[i4];
  const v4f p = ((const v4f*)pb)[i4];
  const float e  = eta[0];
  const float m0 = mu[b * 2 + 0];
  const float m1 = mu[b * 2 + 1];

  v4f r;
#pragma unroll
  for (int j = 0; j < 4; ++j) {
    const float d0 = p[j] - m0;
    const float d1 = p[j] - m1;
    r[j] = z[j] + e * (k[j] + d0 * d0 - d1 * d1);
  }
  __builtin_nontemporal_store(r, (v4f*)out + i4);
}

// ---------------------------------------------------------------------------
// launch
// ---------------------------------------------------------------------------
extern "C" void kernel_launch(void* const* d_in, const int* in_sizes, int n_in,
                              void* d_out, int out_size, void* d_ws,
                              size_t ws_size, hipStream_t stream) {
  (void)in_sizes; (void)n_in; (void)out_size; (void)ws_size;
  const float* kappa = (const float*)d_in[0];
  const float* phi0  = (const float*)d_in[1];
  const float* pb    = (const float*)d_in[2];
  const float* eta   = (const float*)d_in[3];
  // d_in[4] = it (int); unused: eps = 1.0 * 1.0^it == 1.0

  float* partials = (float*)d_ws;                       // NBATCH*R_BLOCKS*3 floats
  float* mu = partials + (size_t)NBATCH * R_BLOCKS * 3; // NBATCH*2 floats
  float* out = (float*)d_out;

  reduce_pass<<<dim3(R_BLOCKS, NBATCH), TPB, 0, stream>>>(phi0, pb, partials);
  mu_pass<<<NBATCH, TPB, 0, stream>>>(partials, mu);
  phi_pass<<<dim3(NPB / (TPB * VEC), NBATCH), TPB, 0, stream>>>(
      kappa, phi0, pb, eta, mu, out);
}